// DGCNConv_12601434046915
// MI455X (gfx1250) — compile-verified
//
#include <hip/hip_runtime.h>
#include <hip/hip_bf16.h>

typedef __attribute__((ext_vector_type(2))) float v2f;
typedef __attribute__((ext_vector_type(8))) float v8f;

#define WPITCH 68   // 68 mod 64 = 4 -> conflict-free LDS float2 reads across wave32
#define BN_EPS 1e-5f

static __device__ __forceinline__ void atomic_add_f32(float* p, float v) {
    __hip_atomic_fetch_add(p, v, __ATOMIC_RELAXED, __HIP_MEMORY_SCOPE_AGENT);
}

// ---------------------------------------------------------------------------
// Edge scatter: 16 threads per edge, each owns a float4 chunk of the 64-float
// row. agg_in[dst] += x[src]; agg_out[src] += x[dst].
// ---------------------------------------------------------------------------
__global__ void dgcn_scatter_kernel(const float* __restrict__ x,
                                    const long long* __restrict__ ei,
                                    float* __restrict__ agg_in,
                                    float* __restrict__ agg_out,
                                    int n_edges) {
    int t = blockIdx.x * blockDim.x + threadIdx.x;
    int e = t >> 4;
    if (e >= n_edges) return;
    int c = (t & 15) << 2;

    int src = (int)ei[e];
    int dst = (int)ei[n_edges + e];

    const float4 vs = *(const float4*)(x + (size_t)src * 64 + c);
    const float4 vd = *(const float4*)(x + (size_t)dst * 64 + c);

    float* pi = agg_in  + (size_t)dst * 64 + c;
    float* po = agg_out + (size_t)src * 64 + c;
    atomic_add_f32(pi + 0, vs.x); atomic_add_f32(pi + 1, vs.y);
    atomic_add_f32(pi + 2, vs.z); atomic_add_f32(pi + 3, vs.w);
    atomic_add_f32(po + 0, vd.x); atomic_add_f32(po + 1, vd.y);
    atomic_add_f32(po + 2, vd.z); atomic_add_f32(po + 3, vd.w);
}

// ---------------------------------------------------------------------------
// Fused triple GEMM + ReLU + BN-stat accumulation.
// out = x@W_self^T + agg_in@W_in^T + agg_out@W_out^T, via V_WMMA_F32_16X16X4_F32.
// One wave32 computes a 16-row x 64-col tile (4 N-tiles of 16).
// ---------------------------------------------------------------------------
__global__ void dgcn_gemm_kernel(const float* __restrict__ x,
                                 const float* __restrict__ agg_in,
                                 const float* __restrict__ agg_out,
                                 const float* __restrict__ W_self,
                                 const float* __restrict__ W_in,
                                 const float* __restrict__ W_out,
                                 float* __restrict__ r_out,
                                 float* __restrict__ stat_sum,
                                 float* __restrict__ stat_sq,
                                 int n_nodes) {
    __shared__ float wlds[3 * 64 * WPITCH];

    const int tid = threadIdx.x;
    const float* Ws[3] = {W_self, W_in, W_out};
#pragma unroll
    for (int s = 0; s < 3; ++s) {
        for (int e = tid; e < 64 * 64; e += 256) {
            int n = e >> 6, k = e & 63;
            wlds[s * 64 * WPITCH + n * WPITCH + k] = Ws[s][e];
        }
    }
    __syncthreads();

    const int wave  = tid >> 5;
    const int lane  = tid & 31;
    const int strip = (blockIdx.x * 8 + wave) * 16;   // N_NODES % 16 == 0
    if (strip >= n_nodes) return;                      // wave-uniform: EXEC stays full

    const int mrow  = lane & 15;      // A row within tile / B col within N-tile
    const int khalf = lane >> 4;      // which K pair (K=0,1 vs K=2,3)

    const float* srcs[3] = {x, agg_in, agg_out};

    v8f acc0 = {}, acc1 = {}, acc2 = {}, acc3 = {};
#pragma unroll
    for (int s = 0; s < 3; ++s) {
        const float* A = srcs[s] + (size_t)(strip + mrow) * 64 + khalf * 2;
        const float* B = &wlds[s * 64 * WPITCH + mrow * WPITCH + khalf * 2];
#pragma unroll
        for (int k0 = 0; k0 < 64; k0 += 4) {
            v2f a  = *(const v2f*)(A + k0);
            v2f b0 = *(const v2f*)(B + k0);
            v2f b1 = *(const v2f*)(B + 16 * WPITCH + k0);
            v2f b2 = *(const v2f*)(B + 32 * WPITCH + k0);
            v2f b3 = *(const v2f*)(B + 48 * WPITCH + k0);
            acc0 = __builtin_amdgcn_wmma_f32_16x16x4_f32(false, a, false, b0,
                                                         (short)0, acc0, false, false);
            acc1 = __builtin_amdgcn_wmma_f32_16x16x4_f32(false, a, false, b1,
                                                         (short)0, acc1, false, false);
            acc2 = __builtin_amdgcn_wmma_f32_16x16x4_f32(false, a, false, b2,
                                                         (short)0, acc2, false, false);
            acc3 = __builtin_amdgcn_wmma_f32_16x16x4_f32(false, a, false, b3,
                                                         (short)0, acc3, false, false);
        }
    }

    // Epilogue: ReLU, store r, accumulate per-feature sum / sum^2.
    // C/D layout: VGPR i -> row (i or i+8 by lane half), col = lane&15 within tile.
    const int rbase = strip + (khalf ? 8 : 0);
    v8f accs[4] = {acc0, acc1, acc2, acc3};
#pragma unroll
    for (int t = 0; t < 4; ++t) {
        const int col = t * 16 + mrow;
        float s1 = 0.f, s2 = 0.f;
#pragma unroll
        for (int i = 0; i < 8; ++i) {
            float v = accs[t][i];
            v = v > 0.f ? v : 0.f;
            r_out[(size_t)(rbase + i) * 64 + col] = v;
            s1 += v;
            s2 += v * v;
        }
        atomic_add_f32(&stat_sum[col], s1);
        atomic_add_f32(&stat_sq[col],  s2);
    }
}

// ---------------------------------------------------------------------------
// BatchNorm apply (in place on d_out). mean/var derived per feature.
// ---------------------------------------------------------------------------
__global__ void dgcn_bn_kernel(float* __restrict__ out,
                               const float* __restrict__ stat_sum,
                               const float* __restrict__ stat_sq,
                               const float* __restrict__ gamma,
                               const float* __restrict__ beta,
                               int n_nodes) {
    long long i = (long long)blockIdx.x * blockDim.x + threadIdx.x;
    long long total = (long long)n_nodes * 64;
    if (i >= total) return;
    int f = (int)(i & 63);
    float invN = 1.0f / (float)n_nodes;
    float mean = stat_sum[f] * invN;
    float var  = stat_sq[f] * invN - mean * mean;
    float r = out[i];
    out[i] = gamma[f] * (r - mean) * rsqrtf(var + BN_EPS) + beta[f];
}

extern "C" void kernel_launch(void* const* d_in, const int* in_sizes, int n_in,
                              void* d_out, int out_size, void* d_ws, size_t ws_size,
                              hipStream_t stream) {
    const float*     x      = (const float*)d_in[0];
    const long long* ei     = (const long long*)d_in[1];
    // d_in[2] = num_nodes scalar (unused; derived from in_sizes)
    const float*     W_in   = (const float*)d_in[3];
    const float*     W_out  = (const float*)d_in[4];
    const float*     W_self = (const float*)d_in[5];
    const float*     gamma  = (const float*)d_in[6];
    const float*     beta   = (const float*)d_in[7];

    const int n_nodes = in_sizes[0] / 64;
    const int n_edges = in_sizes[1] / 2;

    float* agg_in   = (float*)d_ws;
    float* agg_out  = agg_in  + (size_t)n_nodes * 64;
    float* stat_sum = agg_out + (size_t)n_nodes * 64;
    float* stat_sq  = stat_sum + 64;

    // Zero aggregation buffers + BN stat counters (graph-capturable memset node).
    hipMemsetAsync(d_ws, 0, ((size_t)n_nodes * 64 * 2 + 128) * sizeof(float), stream);

    // Edge scatter: 16 threads per edge.
    {
        long long threads = (long long)n_edges * 16;
        int blocks = (int)((threads + 255) / 256);
        dgcn_scatter_kernel<<<blocks, 256, 0, stream>>>(x, ei, agg_in, agg_out, n_edges);
    }

    // Fused GEMM (WMMA f32) + ReLU + BN stats. 8 waves x 16 rows = 128 rows/block.
    {
        int blocks = (n_nodes + 127) / 128;
        dgcn_gemm_kernel<<<blocks, 256, 0, stream>>>(x, agg_in, agg_out,
                                                     W_self, W_in, W_out,
                                                     (float*)d_out, stat_sum, stat_sq,
                                                     n_nodes);
    }

    // BN normalize in place.
    {
        long long total = (long long)n_nodes * 64;
        int blocks = (int)((total + 255) / 256);
        dgcn_bn_kernel<<<blocks, 256, 0, stream>>>((float*)d_out, stat_sum, stat_sq,
                                                   gamma, beta, n_nodes);
    }
}